// GAT_15942918603353
// MI455X (gfx1250) — compile-verified
//
#include <hip/hip_runtime.h>
#include <hip/hip_bf16.h>

// -------- GAT, 2 layers, gfx1250 --------
// N=100000, NFEAT=256, NHID=16, HEADS=8 (Fout1=128), NCLASS=40, E=1.6M (+N self loops)
// All matrix work via V_WMMA_F32_16X16X4_F32 (fp32-exact, matches f32 reference math).
// Edge phase: 2 passes per layer (denom, aggregate) with global f32 atomics.
// softmax max-subtraction omitted: it cancels exactly in alpha = ex/sum(ex) and
// logits here are O(1), so exp() cannot overflow.

typedef __attribute__((ext_vector_type(2))) float v2f;
typedef __attribute__((ext_vector_type(8))) float v8f;

#define NFEAT  256
#define FOUT1  128   // HEADS*NHID
#define NHID   16
#define HEADS  8
#define NCLASS 40
#define NCLSP  48    // padded to 3 WMMA column tiles

__device__ __forceinline__ float lrelu(float v) { return v > 0.0f ? v : 0.2f * v; }

// ---------------- utility kernels ----------------

__global__ void zero_f32(float* __restrict__ p, long long n) {
    long long i = (long long)blockIdx.x * blockDim.x + threadIdx.x;
    if (i < n) p[i] = 0.0f;
}

__global__ void build_w2p(const float* __restrict__ W2, float* __restrict__ W2p) {
    int g = blockIdx.x * blockDim.x + threadIdx.x;          // 128*48
    if (g >= FOUT1 * NCLSP) return;
    int r = g / NCLSP, c = g % NCLSP;
    W2p[g] = (c < NCLASS) ? W2[r * NCLASS + c] : 0.0f;
}

// ---------------- layer-1 GEMM: h1[N,128] = x[N,256] @ W1[256,128] ----------------
// 1 block = 16 rows, 8 waves, wave w computes column tile [16w, 16w+16).

__global__ __launch_bounds__(256) void gemm1_wmma(const float* __restrict__ x,
                                                  const float* __restrict__ W1,
                                                  float* __restrict__ h1) {
    __shared__ float xs[16 * NFEAT];                        // 16 KB, contiguous chunk of x
    const int tid  = threadIdx.x;
    const int row0 = blockIdx.x * 16;

    const float4* src = (const float4*)(x + (size_t)row0 * NFEAT);
    float4* dst = (float4*)xs;
    #pragma unroll
    for (int i = tid; i < (16 * NFEAT) / 4; i += 256) dst[i] = src[i];
    __syncthreads();

    const int lane = tid & 31;
    const int wave = tid >> 5;
    const int col0 = wave * 16;
    const int m    = lane & 15;              // row (A) / col (B) within tile
    const int koff = (lane >> 4) << 1;       // lanes 0-15: K+{0,1}; 16-31: K+{2,3}

    v8f acc = {};
    for (int k = 0; k < NFEAT; k += 4) {
        v2f a, b;
        a.x = xs[m * NFEAT + k + koff];
        a.y = xs[m * NFEAT + k + koff + 1];
        b.x = W1[(size_t)(k + koff)     * FOUT1 + col0 + m];
        b.y = W1[(size_t)(k + koff + 1) * FOUT1 + col0 + m];
        acc = __builtin_amdgcn_wmma_f32_16x16x4_f32(false, a, false, b,
                                                    (short)0, acc, false, false);
    }
    // C layout: VGPR r -> row r (lanes 0-15) / row r+8 (lanes 16-31), col = lane&15
    const int rbase = row0 + ((lane >> 4) << 3);
    const int col   = col0 + m;
    #pragma unroll
    for (int r = 0; r < 8; ++r)
        h1[(size_t)(rbase + r) * FOUT1 + col] = acc[r];
}

// ---------------- layer-2 GEMM: h2[N,40] = hrelu[N,128] @ W2p[128,48] ----------------

__global__ __launch_bounds__(96) void gemm2_wmma(const float* __restrict__ hr,
                                                 const float* __restrict__ W2p,
                                                 float* __restrict__ h2) {
    __shared__ float xs[16 * FOUT1];                        // 8 KB
    const int tid  = threadIdx.x;
    const int row0 = blockIdx.x * 16;

    const float4* src = (const float4*)(hr + (size_t)row0 * FOUT1);
    float4* dst = (float4*)xs;
    #pragma unroll
    for (int i = tid; i < (16 * FOUT1) / 4; i += 96) dst[i] = src[i];
    __syncthreads();

    const int lane = tid & 31;
    const int wave = tid >> 5;                              // 0..2
    const int col0 = wave * 16;
    const int m    = lane & 15;
    const int koff = (lane >> 4) << 1;

    v8f acc = {};
    for (int k = 0; k < FOUT1; k += 4) {
        v2f a, b;
        a.x = xs[m * FOUT1 + k + koff];
        a.y = xs[m * FOUT1 + k + koff + 1];
        b.x = W2p[(k + koff)     * NCLSP + col0 + m];
        b.y = W2p[(k + koff + 1) * NCLSP + col0 + m];
        acc = __builtin_amdgcn_wmma_f32_16x16x4_f32(false, a, false, b,
                                                    (short)0, acc, false, false);
    }
    const int rbase = row0 + ((lane >> 4) << 3);
    const int col   = col0 + m;
    if (col < NCLASS) {
        #pragma unroll
        for (int r = 0; r < 8; ++r)
            h2[(size_t)(rbase + r) * NCLASS + col] = acc[r];
    }
}

// ---------------- attention score kernels ----------------

__global__ void att_scores1(const float* __restrict__ h1,
                            const float* __restrict__ att_src,
                            const float* __restrict__ att_dst,
                            float* __restrict__ a_src, float* __restrict__ a_dst,
                            int n_nodes) {
    long long g = (long long)blockIdx.x * blockDim.x + threadIdx.x;  // n*HEADS + h
    if (g >= (long long)n_nodes * HEADS) return;
    int hd = (int)(g & (HEADS - 1));
    long long n = g >> 3;
    const float* hp = h1 + n * FOUT1 + hd * NHID;
    const float* as = att_src + hd * NHID;
    const float* ad = att_dst + hd * NHID;
    float s = 0.0f, d = 0.0f;
    #pragma unroll
    for (int c = 0; c < NHID; ++c) { float v = hp[c]; s += v * as[c]; d += v * ad[c]; }
    a_src[g] = s; a_dst[g] = d;
}

__global__ void att_scores2(const float* __restrict__ h2,
                            const float* __restrict__ att_src,
                            const float* __restrict__ att_dst,
                            float* __restrict__ a_src, float* __restrict__ a_dst,
                            int n_nodes) {
    int n = blockIdx.x * blockDim.x + threadIdx.x;
    if (n >= n_nodes) return;
    const float* hp = h2 + (size_t)n * NCLASS;
    float s = 0.0f, d = 0.0f;
    #pragma unroll
    for (int c = 0; c < NCLASS; ++c) { float v = hp[c]; s += v * att_src[c]; d += v * att_dst[c]; }
    a_src[n] = s; a_dst[n] = d;
}

// ---------------- edge kernels, layer 1 (8 heads) ----------------

__global__ void edge_denom1(const long long* __restrict__ srcI,
                            const long long* __restrict__ dstI,
                            int E, int n_nodes,
                            const float* __restrict__ a_src,
                            const float* __restrict__ a_dst,
                            float* __restrict__ denom) {
    long long g = (long long)blockIdx.x * blockDim.x + threadIdx.x;
    long long tot = ((long long)E + n_nodes) * HEADS;
    if (g >= tot) return;
    int hd = (int)(g & (HEADS - 1));
    long long e = g >> 3;
    int s, d;
    if (e < E) { s = (int)srcI[e]; d = (int)dstI[e]; } else { s = d = (int)(e - E); }
    float v = expf(lrelu(a_src[(size_t)s * HEADS + hd] + a_dst[(size_t)d * HEADS + hd]));
    atomicAdd(&denom[(size_t)d * HEADS + hd], v);
}

__global__ void edge_agg1(const long long* __restrict__ srcI,
                          const long long* __restrict__ dstI,
                          int E, int n_nodes,
                          const float* __restrict__ a_src,
                          const float* __restrict__ a_dst,
                          const float* __restrict__ denom,
                          const float* __restrict__ h1,
                          float* __restrict__ agg) {
    long long g = (long long)blockIdx.x * blockDim.x + threadIdx.x;
    long long tot = ((long long)E + n_nodes) * HEADS;
    if (g >= tot) return;
    int hd = (int)(g & (HEADS - 1));
    long long e = g >> 3;
    int s, d;
    if (e < E) { s = (int)srcI[e]; d = (int)dstI[e]; } else { s = d = (int)(e - E); }
    float ex = expf(lrelu(a_src[(size_t)s * HEADS + hd] + a_dst[(size_t)d * HEADS + hd]));
    float alpha = ex / (denom[(size_t)d * HEADS + hd] + 1e-16f);
    const float* hp = h1  + (size_t)s * FOUT1 + hd * NHID;
    float*       op = agg + (size_t)d * FOUT1 + hd * NHID;
    #pragma unroll
    for (int c = 0; c < NHID; ++c) atomicAdd(&op[c], alpha * hp[c]);
}

__global__ void relu_bias(const float* __restrict__ agg, const float* __restrict__ b1,
                          float* __restrict__ out, long long n) {
    long long i = (long long)blockIdx.x * blockDim.x + threadIdx.x;
    if (i >= n) return;
    out[i] = fmaxf(agg[i] + b1[i & (FOUT1 - 1)], 0.0f);
}

// ---------------- edge kernels, layer 2 (1 head, 40 channels) ----------------

__global__ void edge_denom2(const long long* __restrict__ srcI,
                            const long long* __restrict__ dstI,
                            int E, int n_nodes,
                            const float* __restrict__ a_src,
                            const float* __restrict__ a_dst,
                            float* __restrict__ denom) {
    long long e = (long long)blockIdx.x * blockDim.x + threadIdx.x;
    long long tot = (long long)E + n_nodes;
    if (e >= tot) return;
    int s, d;
    if (e < E) { s = (int)srcI[e]; d = (int)dstI[e]; } else { s = d = (int)(e - E); }
    atomicAdd(&denom[d], expf(lrelu(a_src[s] + a_dst[d])));
}

__global__ void edge_agg2(const long long* __restrict__ srcI,
                          const long long* __restrict__ dstI,
                          int E, int n_nodes,
                          const float* __restrict__ a_src,
                          const float* __restrict__ a_dst,
                          const float* __restrict__ denom,
                          const float* __restrict__ h2,
                          float* __restrict__ out) {
    long long g = (long long)blockIdx.x * blockDim.x + threadIdx.x;
    long long tot = ((long long)E + n_nodes) * NCLASS;
    if (g >= tot) return;
    int c = (int)(g % NCLASS);
    long long e = g / NCLASS;
    int s, d;
    if (e < E) { s = (int)srcI[e]; d = (int)dstI[e]; } else { s = d = (int)(e - E); }
    float alpha = expf(lrelu(a_src[s] + a_dst[d])) / (denom[d] + 1e-16f);
    atomicAdd(&out[(size_t)d * NCLASS + c], alpha * h2[(size_t)s * NCLASS + c]);
}

__global__ void add_bias2(float* __restrict__ out, const float* __restrict__ b2, long long n) {
    long long i = (long long)blockIdx.x * blockDim.x + threadIdx.x;
    if (i >= n) return;
    out[i] = out[i] + b2[i % NCLASS];
}

// ---------------- launcher ----------------

static inline unsigned cdiv(long long a, long long b) { return (unsigned)((a + b - 1) / b); }

extern "C" void kernel_launch(void* const* d_in, const int* in_sizes, int n_in,
                              void* d_out, int out_size, void* d_ws, size_t ws_size,
                              hipStream_t stream) {
    const float*     x        = (const float*)d_in[0];
    const long long* eidx     = (const long long*)d_in[1];   // int64 [2,E]
    const float*     W1       = (const float*)d_in[2];
    const float*     att_src1 = (const float*)d_in[3];
    const float*     att_dst1 = (const float*)d_in[4];
    const float*     b1       = (const float*)d_in[5];
    const float*     W2       = (const float*)d_in[6];
    const float*     att_src2 = (const float*)d_in[7];
    const float*     att_dst2 = (const float*)d_in[8];
    const float*     b2       = (const float*)d_in[9];

    const int N = in_sizes[0] / NFEAT;       // 100000 (divisible by 16)
    const int E = in_sizes[1] / 2;           // 1600000
    const long long Etot = (long long)E + N;
    const long long* srcI = eidx;
    const long long* dstI = eidx + E;

    // workspace layout (floats); total ~130 MB
    float* ws    = (float*)d_ws;
    float* h1    = ws;                               // N*128 (later reused as h_relu)
    float* agg1  = h1    + (size_t)N * FOUT1;        // N*128
    float* asrc1 = agg1  + (size_t)N * FOUT1;        // N*8
    float* adst1 = asrc1 + (size_t)N * HEADS;        // N*8
    float* den1  = adst1 + (size_t)N * HEADS;        // N*8
    float* h2    = den1  + (size_t)N * HEADS;        // N*40
    float* asrc2 = h2    + (size_t)N * NCLASS;       // N
    float* adst2 = asrc2 + (size_t)N;                // N
    float* den2  = adst2 + (size_t)N;                // N
    float* W2p   = den2  + (size_t)N;                // 128*48
    float* out   = (float*)d_out;

    const int B = 256;

    // init accumulators (d_out/d_ws are poisoned by the harness)
    zero_f32<<<cdiv((long long)N * FOUT1, B), B, 0, stream>>>(agg1, (long long)N * FOUT1);
    zero_f32<<<cdiv((long long)N * HEADS, B), B, 0, stream>>>(den1, (long long)N * HEADS);
    zero_f32<<<cdiv(N, B), B, 0, stream>>>(den2, N);
    zero_f32<<<cdiv((long long)N * NCLASS, B), B, 0, stream>>>(out, (long long)N * NCLASS);
    build_w2p<<<cdiv(FOUT1 * NCLSP, B), B, 0, stream>>>(W2, W2p);

    // ---- layer 1 ----
    gemm1_wmma<<<N / 16, 256, 0, stream>>>(x, W1, h1);
    att_scores1<<<cdiv((long long)N * HEADS, B), B, 0, stream>>>(h1, att_src1, att_dst1,
                                                                 asrc1, adst1, N);
    edge_denom1<<<cdiv(Etot * HEADS, B), B, 0, stream>>>(srcI, dstI, E, N, asrc1, adst1, den1);
    edge_agg1<<<cdiv(Etot * HEADS, B), B, 0, stream>>>(srcI, dstI, E, N, asrc1, adst1, den1,
                                                       h1, agg1);
    relu_bias<<<cdiv((long long)N * FOUT1, B), B, 0, stream>>>(agg1, b1, h1,
                                                               (long long)N * FOUT1);

    // ---- layer 2 ----
    gemm2_wmma<<<N / 16, 96, 0, stream>>>(h1, W2p, h2);
    att_scores2<<<cdiv(N, B), B, 0, stream>>>(h2, att_src2, att_dst2, asrc2, adst2, N);
    edge_denom2<<<cdiv(Etot, B), B, 0, stream>>>(srcI, dstI, E, N, asrc2, adst2, den2);
    edge_agg2<<<cdiv(Etot * NCLASS, B), B, 0, stream>>>(srcI, dstI, E, N, asrc2, adst2, den2,
                                                        h2, out);
    add_bias2<<<cdiv((long long)N * NCLASS, B), B, 0, stream>>>(out, b2, (long long)N * NCLASS);
}